// MoeRl_86449101734487
// MI455X (gfx1250) — compile-verified
//
#include <hip/hip_runtime.h>
#include <hip/hip_bf16.h>

// Problem constants (from reference)
#define BB 512
#define SS 196
#define TT 768
#define EE 16
#define KK 4
#define HH 32

// fc1 TDM tiling
#define KC   32              // k-chunk (floats)
#define KCP  34              // padded LDS row stride (floats): conflict-free
#define ROWS 128             // rows per workgroup (8 waves x 16)

typedef __attribute__((ext_vector_type(2))) float v2f;
typedef __attribute__((ext_vector_type(8))) float v8f;
typedef __attribute__((ext_vector_type(4))) unsigned u32x4;
typedef __attribute__((ext_vector_type(8))) int i32x8;
typedef __attribute__((ext_vector_type(4))) int i32x4;

// ---------------------------------------------------------------------------
// Tensor Data Mover: 2D tile load Global -> LDS.
// D# built per CDNA5 ISA 8.3/8.4: group0 = {flags, lds_addr, global_addr, type},
// group1 = {mask/data_size/pad, dims, tile dims, dim0 stride}.
// Padding: after every 32 dwords insert 2 dwords => LDS row stride = 34 floats.
// ---------------------------------------------------------------------------
__device__ __forceinline__ void tdm_load_2d(unsigned ldsAddr, unsigned long long gaddr,
                                            unsigned tensorDim0, unsigned tensorDim1,
                                            unsigned tileDim0, unsigned tileDim1,
                                            unsigned long long stride0) {
  u32x4 g0;
  g0[0] = 1u;                                            // count=1 (valid user D#)
  g0[1] = ldsAddr;                                       // lds_addr (bytes)
  g0[2] = (unsigned)(gaddr & 0xffffffffull);             // global_addr[31:0]
  g0[3] = (unsigned)((gaddr >> 32) & 0x1ffffffull)       // global_addr[56:32]
        | (2u << 30);                                    // type = 2 ("image")
  i32x8 g1;
  g1[0] = (int)((2u << 16)                               // data_size = 4 bytes
        | (1u << 20)                                     // pad_enable
        | (4u << 22)                                     // pad_interval: 2<<4 = 32 dwords
        | (1u << 25));                                   // pad_amount: 2 dwords
  g1[1] = (int)((tensorDim0 & 0xffffu) << 16);           // tensor_dim0 lo16 @ [63:48]
  g1[2] = (int)((tensorDim0 >> 16) | ((tensorDim1 & 0xffffu) << 16)); // dim0 hi / dim1 lo
  g1[3] = (int)((tensorDim1 >> 16) | (tileDim0 << 16));  // dim1 hi16, tile_dim0
  g1[4] = (int)(tileDim1 & 0xffffu);                     // tile_dim1 (tile_dim2 = 0)
  g1[5] = (int)(stride0 & 0xffffffffull);                // tensor_dim0_stride lo32
  g1[6] = (int)((stride0 >> 32) & 0xffffull);            // stride hi16 (dim1_stride = 0)
  g1[7] = 0;
  i32x4 z4 = {0, 0, 0, 0};
#if __clang_major__ >= 23
  i32x8 z8 = {0, 0, 0, 0, 0, 0, 0, 0};
  __builtin_amdgcn_tensor_load_to_lds(g0, g1, z4, z4, z8, 0);
#else
  __builtin_amdgcn_tensor_load_to_lds(g0, g1, z4, z4, 0);
#endif
}

// ---------------------------------------------------------------------------
// fc1: h = x @ fc1_w.T + b   (M=100352, N=32, K=768), f32 WMMA 16x16x4.
// Workgroup = 256 threads (8 waves), 128 rows. x tiles staged to LDS by the
// TDM (double buffered, wave 0 issues + waits TENSORcnt, barrier hands off).
// A-frag from LDS at padded stride (conflict-free); W rows from global (L0-hot).
// ---------------------------------------------------------------------------
__global__ void __launch_bounds__(256)
fc1_wmma_tdm(const float* __restrict__ x, const float* __restrict__ w,
             const float* __restrict__ bias, float* __restrict__ out) {
  __shared__ float lds[2 * ROWS * KCP];

  const int wave = threadIdx.x >> 5;
  const int lane = threadIdx.x & 31;
  const int lq   = lane & 15;
  const int half = lane >> 4;
  const long rowBase = (long)blockIdx.x * ROWS;

  const float* w0 = w + (long)lq * TT + 2 * half;         // n-tile 0 (cols 0..15)
  const float* w1 = w + (long)(16 + lq) * TT + 2 * half;  // n-tile 1 (cols 16..31)

  v8f acc0 = {0.f, 0.f, 0.f, 0.f, 0.f, 0.f, 0.f, 0.f};
  v8f acc1 = {0.f, 0.f, 0.f, 0.f, 0.f, 0.f, 0.f, 0.f};

  const unsigned long long xbase =
      (unsigned long long)(uintptr_t)x + (unsigned long long)(rowBase * TT) * 4ull;
  const unsigned ldsBase = (unsigned)(uintptr_t)(&lds[0]);

  const int nch = TT / KC;  // 24 chunks

  if (wave == 0) {  // prologue: DMA chunk 0 into buffer 0
    tdm_load_2d(ldsBase, xbase, TT, ROWS, KC, ROWS, TT);
  }

  for (int c = 0; c < nch; ++c) {
    if (wave == 0) __builtin_amdgcn_s_wait_tensorcnt(0);
    __syncthreads();  // chunk c visible to all waves; buf (c+1)&1 free
    if (wave == 0 && c + 1 < nch) {
      tdm_load_2d(ldsBase + (unsigned)(((c + 1) & 1) * ROWS * KCP * 4),
                  xbase + (unsigned long long)((c + 1) * KC) * 4ull,
                  TT, ROWS, KC, ROWS, TT);
    }

    const float* la = &lds[(c & 1) * ROWS * KCP + (wave * 16 + lq) * KCP + 2 * half];
    const float* wa = w0 + c * KC;
    const float* wb = w1 + c * KC;
#pragma unroll
    for (int ks = 0; ks < KC; ks += 4) {
      v2f a  = *(const v2f*)(la + ks);
      v2f b0 = *(const v2f*)(wa + ks);
      v2f b1 = *(const v2f*)(wb + ks);
      acc0 = __builtin_amdgcn_wmma_f32_16x16x4_f32(false, a, false, b0,
                                                   (short)0, acc0, false, false);
      acc1 = __builtin_amdgcn_wmma_f32_16x16x4_f32(false, a, false, b1,
                                                   (short)0, acc1, false, false);
    }
    __syncthreads();  // all reads of buf c&1 done before it is DMA'd again
  }

  const float bv0 = bias[lq];
  const float bv1 = bias[16 + lq];
#pragma unroll
  for (int i = 0; i < 8; ++i) {
    const long m = rowBase + wave * 16 + i + 8 * half;
    out[m * HH + lq]      = acc0[i] + bv0;
    out[m * HH + 16 + lq] = acc1[i] + bv1;
  }
}

// ---------------------------------------------------------------------------
// Generic WMMA f32 GEMM: C[z] = act(A[z] @ W[z]^T + bias[z]); W is (N,K).
// One wave per 16x16 C tile. N-tail handled with clamped load + select
// (no exec-mask branches in the inner loop).
// ---------------------------------------------------------------------------
__global__ void wmma_gemm_f32(const float* __restrict__ A, long lda, long aBatch,
                              const float* __restrict__ W, long wBatch,
                              const float* __restrict__ bias, long bBatch,
                              float* __restrict__ C, long ldc, long cBatch,
                              int M, int N, int K, int tilesM, int doRelu) {
  const int z = blockIdx.z;
  A += (long)z * aBatch;
  W += (long)z * wBatch;
  C += (long)z * cBatch;
  if (bias) bias += (long)z * bBatch;

  const int wavesPerBlk = blockDim.x >> 5;
  const int waveInBlk   = threadIdx.x >> 5;
  const int lane        = threadIdx.x & 31;
  const int mtile       = blockIdx.y * wavesPerBlk + waveInBlk;
  const int ntile       = blockIdx.x;
  if (mtile >= tilesM) return;  // wave-uniform

  const int lq   = lane & 15;
  const int half = lane >> 4;

  const long rowA = (long)mtile * 16 + lq;
  const int  colW = ntile * 16 + lq;
  const bool nOK  = (colW < N);
  const int  colWc = nOK ? colW : (N - 1);     // clamp: always-legal load
  const float* aRow = A + rowA * lda + 2 * half;
  const float* wRow = W + (long)colWc * (long)K + 2 * half;

  v8f acc = {0.f, 0.f, 0.f, 0.f, 0.f, 0.f, 0.f, 0.f};

#pragma unroll 4
  for (int k0 = 0; k0 < K; k0 += 4) {
    v2f a = *(const v2f*)(aRow + k0);
    v2f b = *(const v2f*)(wRow + k0);
    if (!nOK) { b[0] = 0.f; b[1] = 0.f; }      // v_cndmask, no branch
    acc = __builtin_amdgcn_wmma_f32_16x16x4_f32(false, a, false, b,
                                                (short)0, acc, false, false);
  }

  const int n = ntile * 16 + lq;
  float bv = 0.f;
  if (bias && nOK) bv = bias[colWc];

#pragma unroll
  for (int i = 0; i < 8; ++i) {
    const int m = mtile * 16 + i + 8 * half;
    float v = acc[i] + bv;
    if (doRelu) v = fmaxf(v, 0.f);
    if (nOK) C[(long)m * ldc + n] = v;
  }
}

// ---------------------------------------------------------------------------
// JAX threefry2x32 (key (0,42)) matching jax.random.uniform over (B,E,S).
// ---------------------------------------------------------------------------
__device__ __forceinline__ unsigned rotl32(unsigned x, int r) {
  return (x << r) | (x >> (32 - r));
}

__device__ float jax_uniform42(unsigned i, unsigned halfN) {
  unsigned c0, c1; int sel;
  if (i < halfN) { c0 = i;         c1 = i + halfN; sel = 0; }
  else           { c0 = i - halfN; c1 = i;         sel = 1; }
  const unsigned k0 = 0u, k1 = 42u;
  const unsigned ks[3] = {k0, k1, k0 ^ k1 ^ 0x1BD11BDAu};
  unsigned x0 = c0 + ks[0], x1 = c1 + ks[1];
  const int rots[2][4] = {{13, 15, 26, 6}, {17, 29, 16, 24}};
#pragma unroll
  for (int blk = 0; blk < 5; ++blk) {
#pragma unroll
    for (int j = 0; j < 4; ++j) {
      x0 += x1;
      x1 = rotl32(x1, rots[blk & 1][j]);
      x1 ^= x0;
    }
    x0 += ks[(blk + 1) % 3];
    x1 += ks[(blk + 2) % 3] + (unsigned)(blk + 1);
  }
  unsigned bits = sel ? x1 : x0;
  float f = __uint_as_float((bits >> 9) | 0x3F800000u) - 1.0f;
  const float mn = 1e-6f, mx = 1.0f - 1e-6f;
  return fmaxf(f * (mx - mn) + mn, mn);
}

// ---------------------------------------------------------------------------
// Per (b,e): softmax over S of logits[b,:,e], logp + Gumbel, top-4 indices.
// One wave32 per (b,e).
// ---------------------------------------------------------------------------
__global__ void softmax_gumbel_topk(const float* __restrict__ logits,
                                    int* __restrict__ idxOut) {
  const int gwave = (int)((blockIdx.x * blockDim.x + threadIdx.x) >> 5);
  const int lane  = (int)(threadIdx.x & 31);
  if (gwave >= BB * EE) return;
  const int b = gwave >> 4;
  const int e = gwave & 15;

  const float NEG = -3.0e38f;
  float v[7];
#pragma unroll
  for (int j = 0; j < 7; ++j) {
    int s = lane + 32 * j;
    v[j] = (s < SS) ? logits[((long)b * SS + s) * EE + e] : NEG;
  }

  float mx = v[0];
#pragma unroll
  for (int j = 1; j < 7; ++j) mx = fmaxf(mx, v[j]);
  for (int off = 16; off; off >>= 1) mx = fmaxf(mx, __shfl_xor(mx, off, 32));

  float se = 0.f;
#pragma unroll
  for (int j = 0; j < 7; ++j) {
    int s = lane + 32 * j;
    if (s < SS) se += expf(v[j] - mx);
  }
  for (int off = 16; off; off >>= 1) se += __shfl_xor(se, off, 32);
  const float lse = logf(se);

  const unsigned halfN = (unsigned)(BB * EE * SS) / 2u;
  float sc[7];
#pragma unroll
  for (int j = 0; j < 7; ++j) {
    int s = lane + 32 * j;
    if (s < SS) {
      unsigned flat = ((unsigned)(b * EE + e)) * (unsigned)SS + (unsigned)s;
      float u = jax_uniform42(flat, halfN);
      float g = -logf(-logf(u));
      sc[j] = (v[j] - mx - lse) + g;
    } else {
      sc[j] = NEG;
    }
  }

#pragma unroll
  for (int kk = 0; kk < KK; ++kk) {
    float bv = NEG;
    int bs = 0x7fffffff;
#pragma unroll
    for (int j = 0; j < 7; ++j) {
      int s = lane + 32 * j;
      if (sc[j] > bv || (sc[j] == bv && s < bs)) { bv = sc[j]; bs = s; }
    }
    for (int off = 16; off; off >>= 1) {
      float ov = __shfl_xor(bv, off, 32);
      int   os = __shfl_xor(bs, off, 32);
      if (ov > bv || (ov == bv && os < bs)) { bv = ov; bs = os; }
    }
    if (lane == 0) idxOut[gwave * KK + kk] = bs;
#pragma unroll
    for (int j = 0; j < 7; ++j) {
      if (lane + 32 * j == bs) sc[j] = NEG;
    }
  }
}

// ---------------------------------------------------------------------------
// inp[b, e, k*32 + h] = hbuf[b, idx[b,e,k], h]   (flat tid == inp index)
// ---------------------------------------------------------------------------
__global__ void gather_inp(const float* __restrict__ hbuf,
                           const int* __restrict__ idx,
                           float* __restrict__ inp) {
  const long t = (long)blockIdx.x * blockDim.x + threadIdx.x;
  if (t >= (long)BB * EE * KK * HH) return;
  const int hh = (int)(t & 31);
  const int kk = (int)((t >> 5) & 3);
  const int e  = (int)((t >> 7) & 15);
  const int b  = (int)(t >> 11);
  const int s  = idx[(b * EE + e) * KK + kk];
  inp[t] = hbuf[((long)b * SS + s) * HH + hh];
}

// ---------------------------------------------------------------------------
extern "C" void kernel_launch(void* const* d_in, const int* in_sizes, int n_in,
                              void* d_out, int out_size, void* d_ws, size_t ws_size,
                              hipStream_t stream) {
  const float* x      = (const float*)d_in[0];
  const float* fc1_w  = (const float*)d_in[1];
  const float* fc1_b  = (const float*)d_in[2];
  const float* gate_w = (const float*)d_in[3];
  const float* gate_b = (const float*)d_in[4];
  const float* exp_w  = (const float*)d_in[5];
  const float* exp_b  = (const float*)d_in[6];
  const float* fc2_w  = (const float*)d_in[7];
  const float* fc2_b  = (const float*)d_in[8];
  const float* fc3_w  = (const float*)d_in[9];
  const float* fc3_b  = (const float*)d_in[10];
  const float* fc4_w  = (const float*)d_in[11];
  const float* fc4_b  = (const float*)d_in[12];
  const float* fc5_w  = (const float*)d_in[13];
  const float* fc5_b  = (const float*)d_in[14];
  const float* fc6_w  = (const float*)d_in[15];
  const float* fc6_b  = (const float*)d_in[16];
  float* out = (float*)d_out;

  // Workspace layout (floats) -- ~33.5 MB (32 MiB) total
  float* ws     = (float*)d_ws;
  float* hbuf   = ws;                                  // B*S*H   = 3,211,264
  float* logits = hbuf + (long)BB * SS * HH;           // B*S*E   = 1,605,632
  int*   idx    = (int*)(logits + (long)BB * SS * EE); // B*E*K   =    32,768
  float* inp    = (float*)(idx + (long)BB * EE * KK);  // B*E*K*H = 1,048,576
  float* o1     = inp + (long)BB * EE * KK * HH;       // B*E*32  =   262,144
  float* o2     = o1  + (long)BB * EE * 32;            // B*E*128 = 1,048,576
  float* o3     = o2  + (long)BB * EE * 128;           // B*E*128 = 1,048,576
  float* o4     = o3  + (long)BB * EE * 128;           // B*128   =    65,536
  float* o5     = o4  + (long)BB * 128;                // B*128   =    65,536

  const int BLK = 128;       // 4 waves per block (generic GEMM)
  const int M1  = BB * SS;   // 100352 rows
  const int T1  = M1 / 16;   // 6272 M-tiles

  // fc1 (HBM-bound, 308 MB stream): TDM-staged WMMA GEMM, 128 rows/block
  fc1_wmma_tdm<<<dim3(M1 / ROWS), 256, 0, stream>>>(x, fc1_w, fc1_b, hbuf);

  // gate: logits = h @ gate_w.T + b    (M=100352, N=16, K=32)
  wmma_gemm_f32<<<dim3(1, T1 / 4, 1), BLK, 0, stream>>>(
      hbuf, HH, 0, gate_w, 0, gate_b, 0, logits, EE, 0, M1, EE, HH, T1, 0);

  // softmax over S + gumbel + top-4 per (b,e): 8192 waves
  softmax_gumbel_topk<<<dim3((BB * EE * 32) / 256), 256, 0, stream>>>(logits, idx);

  // gather selected rows of h into inp (B,E,128)
  gather_inp<<<dim3((BB * EE * KK * HH) / 256), 256, 0, stream>>>(hbuf, idx, inp);

  // expert einsum: per-e GEMM (M=512, N=32, K=128), batched over z=E, +relu
  wmma_gemm_f32<<<dim3(2, 8, EE), BLK, 0, stream>>>(
      inp, (long)EE * 128, 128, exp_w, 32L * 128, exp_b, 32,
      o1, (long)EE * 32, 32, BB, 32, 128, BB / 16, 1);

  // fc2: (8192,32) -> (8192,128), relu
  wmma_gemm_f32<<<dim3(8, 128, 1), BLK, 0, stream>>>(
      o1, 32, 0, fc2_w, 0, fc2_b, 0, o2, 128, 0, BB * EE, 128, 32, (BB * EE) / 16, 1);

  // fc3: (8192,128) -> (8192,128), relu
  wmma_gemm_f32<<<dim3(8, 128, 1), BLK, 0, stream>>>(
      o2, 128, 0, fc3_w, 0, fc3_b, 0, o3, 128, 0, BB * EE, 128, 128, (BB * EE) / 16, 1);

  // fc4: (512,2048) -> (512,128), relu
  wmma_gemm_f32<<<dim3(8, 8, 1), BLK, 0, stream>>>(
      o3, 2048, 0, fc4_w, 0, fc4_b, 0, o4, 128, 0, BB, 128, 2048, BB / 16, 1);

  // fc5: (512,128) -> (512,128), relu
  wmma_gemm_f32<<<dim3(8, 8, 1), BLK, 0, stream>>>(
      o4, 128, 0, fc5_w, 0, fc5_b, 0, o5, 128, 0, BB, 128, 128, BB / 16, 1);

  // fc6: (512,128) -> (512,10)
  wmma_gemm_f32<<<dim3(1, 8, 1), BLK, 0, stream>>>(
      o5, 128, 0, fc6_w, 0, fc6_b, 0, out, 10, 0, BB, 10, 128, BB / 16, 0);
}